// ComplexConv2Deffangle4Dxy_89541478187134
// MI455X (gfx1250) — compile-verified
//
#include <hip/hip_runtime.h>
#include <cmath>

typedef __attribute__((ext_vector_type(2))) float v2f;
typedef __attribute__((ext_vector_type(8))) float v8f;

#define EPSV 1e-6f

// ---------------------------------------------------------------------------
// Weight normalization prep: w1n = (w1x+w1y)^2 / sum, w2n = (w2x+w2y)^2 / sum
// Single block, deterministic tree reduction.
// ---------------------------------------------------------------------------
__global__ __launch_bounds__(256) void wnorm_prep(
    const float* __restrict__ w1x, const float* __restrict__ w1y,
    const float* __restrict__ w2x, const float* __restrict__ w2y,
    float* __restrict__ w1n, float* __restrict__ w2n) {
  __shared__ float red[256];
  const int tid = threadIdx.x;

  // ---- w1n : 64*9 = 576 elements ----
  float s = 0.f;
  for (int i = tid; i < 576; i += 256) { float v = w1x[i] + w1y[i]; s += v * v; }
  red[tid] = s;
  __syncthreads();
  for (int off = 128; off > 0; off >>= 1) {
    if (tid < off) red[tid] += red[tid + off];
    __syncthreads();
  }
  const float inv1 = 1.0f / red[0];
  __syncthreads();
  for (int i = tid; i < 576; i += 256) { float v = w1x[i] + w1y[i]; w1n[i] = v * v * inv1; }

  // ---- w2n : 128*64 = 8192 elements ----
  s = 0.f;
  for (int i = tid; i < 8192; i += 256) { float v = w2x[i] + w2y[i]; s += v * v; }
  __syncthreads();
  red[tid] = s;
  __syncthreads();
  for (int off = 128; off > 0; off >>= 1) {
    if (tid < off) red[tid] += red[tid + off];
    __syncthreads();
  }
  const float inv2 = 1.0f / red[0];
  __syncthreads();
  for (int i = tid; i < 8192; i += 256) { float v = w2x[i] + w2y[i]; w2n[i] = v * v * inv2; }
}

// ---------------------------------------------------------------------------
// Fused depthwise-3x3 + 128x64 GEMM (WMMA f32 16x16x4), one output row/block.
// Grid: (64 rows, 4 branches, 8 batch). Block: 256 threads = 8 waves.
// ---------------------------------------------------------------------------
__global__ __launch_bounds__(256) void ComplexConv2Deffangle4Dxy_kernel(
    const float* __restrict__ x,
    const float* __restrict__ w1x, const float* __restrict__ w1y,
    const float* __restrict__ w2x, const float* __restrict__ w2y,
    const float* __restrict__ w1n, const float* __restrict__ w2n,
    float* __restrict__ out) {
  constexpr int Cin = 64, Cout = 128, Hd = 64, Wd = 64;
  constexpr int TSTR = 65;  // LDS row stride (bank-conflict padding)
  __shared__ float t_lds[Cin * TSTR];

  const int y = blockIdx.x;  // output row 0..63
  const int f = blockIdx.y;  // branch 0..3
  const int b = blockIdx.z;  // batch 0..7

  const float* w1;
  const float* w2;
  if (f < 2)      { w1 = w1n; w2 = w2n; }
  else if (f == 2){ w1 = w1x; w2 = w2x; }
  else            { w1 = w1y; w2 = w2y; }
  const bool is_log = (f == 1);

  const float* __restrict__ xin = x + (size_t)(b * 4 + f) * Cin * Hd * Wd;

  const int tid = threadIdx.x;

  // ----------------- Phase 1: depthwise 3x3 conv -> t_lds[c][p] -------------
  {
    const int c  = tid >> 2;        // channel 0..63
    const int p0 = (tid & 3) * 16;  // pixel group start 0/16/32/48

    float wk[9];
#pragma unroll
    for (int k = 0; k < 9; ++k) wk[k] = w1[c * 9 + k];

    float rb[3][18];  // 3 input rows, 18 columns (16 + halo)
#pragma unroll
    for (int dy = 0; dy < 3; ++dy) {
      const int ry = y + dy - 1;
      const bool rvalid = (ry >= 0) && (ry < Hd);
      const float* __restrict__ rp = xin + (size_t)c * Hd * Wd + (size_t)(rvalid ? ry : 0) * Wd;
#pragma unroll
      for (int j = 0; j < 18; ++j) {
        const int cx = p0 - 1 + j;
        float v = 0.0f;
        if (rvalid && cx >= 0 && cx < Wd) v = rp[cx];
        if (is_log) v = __logf(v + EPSV);  // padding -> log(EPS), as reference
        rb[dy][j] = v;
      }
    }
#pragma unroll
    for (int i = 0; i < 16; ++i) {
      float acc = 0.0f;
#pragma unroll
      for (int dy = 0; dy < 3; ++dy)
#pragma unroll
        for (int dx = 0; dx < 3; ++dx)
          acc = fmaf(rb[dy][i + dx], wk[dy * 3 + dx], acc);
      t_lds[c * TSTR + p0 + i] = acc;
    }
  }
  __syncthreads();

  // ----------------- Phase 2: out[o,p] = sum_c w2[o,c] * t[c,p] via WMMA ---
  const int wave = tid >> 5;       // 0..7  (wave32)
  const int lane = tid & 31;
  const int o_base = wave * 16;
  const int mrow   = lane & 15;        // M (A) / N (B) index within tile
  const int koff   = (lane >> 4) * 2;  // K sub-offset: lanes 0-15 -> {0,1}, 16-31 -> {2,3}

  // Preload A fragments: A = w2 tile [16 x 4] per step, 16 steps cover K=64.
  v2f a[16];
  const float* __restrict__ w2row = w2 + (size_t)(o_base + mrow) * Cin;
#pragma unroll
  for (int s = 0; s < 16; ++s) {
    a[s].x = w2row[4 * s + koff + 0];
    a[s].y = w2row[4 * s + koff + 1];
  }

  float* __restrict__ outp =
      out + (size_t)(b * 4 + f) * Cout * (Hd * Wd) + (size_t)y * Wd;

  for (int pt = 0; pt < 4; ++pt) {
    const int p_base = pt * 16;
    const int n = p_base + mrow;

    v8f acc = {0.f, 0.f, 0.f, 0.f, 0.f, 0.f, 0.f, 0.f};
#pragma unroll
    for (int s = 0; s < 16; ++s) {
      v2f bf;
      bf.x = t_lds[(4 * s + koff + 0) * TSTR + n];
      bf.y = t_lds[(4 * s + koff + 1) * TSTR + n];
      acc = __builtin_amdgcn_wmma_f32_16x16x4_f32(
          /*neg_a=*/false, a[s], /*neg_b=*/false, bf,
          /*c_mod=*/(short)0, acc, /*reuse_a=*/false, /*reuse_b=*/false);
    }

    // Store C/D tile: VGPR i -> M = i (lanes 0-15) or M = 8+i (lanes 16-31)
    const int col   = n;                          // output column (x)
    const int orow0 = o_base + (lane >> 4) * 8;   // +8 for upper lane half
#pragma unroll
    for (int i = 0; i < 8; ++i) {
      float v = acc[i];
      if (is_log) v = __expf(v);
      outp[(size_t)(orow0 + i) * (Hd * Wd) + col] = v;
    }
  }
}

// ---------------------------------------------------------------------------
extern "C" void kernel_launch(void* const* d_in, const int* in_sizes, int n_in,
                              void* d_out, int out_size, void* d_ws, size_t ws_size,
                              hipStream_t stream) {
  const float* x   = (const float*)d_in[0];
  const float* w1x = (const float*)d_in[1];
  const float* w1y = (const float*)d_in[2];
  const float* w2x = (const float*)d_in[3];
  const float* w2y = (const float*)d_in[4];

  float* w1n = (float*)d_ws;       // 576 floats
  float* w2n = w1n + 576;          // 8192 floats
  float* out = (float*)d_out;

  wnorm_prep<<<1, 256, 0, stream>>>(w1x, w1y, w2x, w2y, w1n, w2n);

  dim3 grid(64 /*rows*/, 4 /*branch*/, 8 /*batch*/);
  ComplexConv2Deffangle4Dxy_kernel<<<grid, 256, 0, stream>>>(
      x, w1x, w1y, w2x, w2y, w1n, w2n, out);
}